// color_loss_3813930958865
// MI455X (gfx1250) — compile-verified
//
#include <hip/hip_runtime.h>

typedef __attribute__((ext_vector_type(2))) float v2f;
typedef __attribute__((ext_vector_type(4))) float v4f;
typedef __attribute__((ext_vector_type(8))) float v8f;

#define EPS_F   1e-6f
#define HW      262144      // 512*512
#define CHW     786432      // 3*HW
#define NPIX    4194304     // 16*HW
#define QUADS   1048576     // NPIX/4
#define GRID1   1024
#define BLOCK   256
#define NWAVES  (BLOCK / 32)
#define NTHR    (GRID1 * BLOCK)
#define ITERS   (QUADS / NTHR)      // exactly 4
#define PI_D    3.14159265358979323846
#define SWAPX16 0x401F              // ds_swizzle group-of-32: xor=0x10, or=0, and=0x1f

// Non-temporal streaming load (data is touched exactly once -> TH_NT).
__device__ __forceinline__ v4f ldg_nt(const float* p) {
    return __builtin_nontemporal_load((const v4f*)p);
}

// Full 32-lane (wave32) sum via V_WMMA_F32_16X16X4_F32.
// A: VGPR0 = per-lane partial, VGPR1 = 0  -> A[m][0]=s_m, A[m][2]=s_{m+16}
// B: all ones (4x16)                      -> D[m][n] = s_m + s_{m+16}
// Lane L<16 holds D[0..7][L] in its 8 D VGPRs; lane L>=16 holds D[8..15][L-16].
// Sum own 8 D VGPRs, then single ds_swizzle SWAPX16 folds the two halves.
__device__ __forceinline__ float wave_sum_wmma(float v) {
    v2f a; a[0] = v;    a[1] = 0.0f;
    v2f b; b[0] = 1.0f; b[1] = 1.0f;
    v8f c = {};
    v8f d = __builtin_amdgcn_wmma_f32_16x16x4_f32(
        /*neg_a=*/false, a, /*neg_b=*/false, b,
        /*c_mod=*/(short)0, c, /*reuse_a=*/false, /*reuse_b=*/false);
    float s = ((d[0] + d[1]) + (d[2] + d[3])) + ((d[4] + d[5]) + (d[6] + d[7]));
    s += __int_as_float(__builtin_amdgcn_ds_swizzle(__float_as_int(s), SWAPX16));
    return s;
}

__device__ __forceinline__ float pixel_angle(float xr, float xg, float xb,
                                             float yr, float yg, float yb) {
    xr += EPS_F; xg += EPS_F; xb += EPS_F;
    yr += EPS_F; yg += EPS_F; yb += EPS_F;
    float lx2 = xr * xr + xg * xg + xb * xb;
    float ly2 = yr * yr + yg * yg + yb * yb;
    float xy  = xr * yr + xg * yg + xb * yb;
    float cosv = xy / (sqrtf(lx2) * sqrtf(ly2) + EPS_F);
    cosv = fminf(fmaxf(cosv, -1.0f), 1.0f);
    return acosf(cosv);   // radians; degree conversion deferred to finalize
}

__global__ void __launch_bounds__(BLOCK)
angle_partial_kernel(const float* __restrict__ x, const float* __restrict__ y,
                     float* __restrict__ ws) {
    const int tid = blockIdx.x * BLOCK + threadIdx.x;

    // Fixed trip count: no exec-masked loop, all loads visible for pipelining.
    v4f xv[ITERS][3], yv[ITERS][3];
#pragma unroll
    for (int k = 0; k < ITERS; ++k) {
        const int  q    = tid + k * NTHR;
        const int  p    = q << 2;          // global pixel index = b*HW + i
        const int  bb   = p >> 18;         // / HW (power of two)
        const int  i    = p & (HW - 1);    // % HW
        const long base = (long)bb * CHW + i;
        xv[k][0] = ldg_nt(x + base);
        xv[k][1] = ldg_nt(x + base + HW);
        xv[k][2] = ldg_nt(x + base + 2 * HW);
        yv[k][0] = ldg_nt(y + base);
        yv[k][1] = ldg_nt(y + base + HW);
        yv[k][2] = ldg_nt(y + base + 2 * HW);
    }

    float acc = 0.0f;
#pragma unroll
    for (int k = 0; k < ITERS; ++k) {
#pragma unroll
        for (int j = 0; j < 4; ++j) {
            acc += pixel_angle(xv[k][0][j], xv[k][1][j], xv[k][2][j],
                               yv[k][0][j], yv[k][1][j], yv[k][2][j]);
        }
    }

    __shared__ float smem[NWAVES];
    const float wsum = wave_sum_wmma(acc);   // EXEC all-ones: uniform control flow
    if ((threadIdx.x & 31) == 0) smem[threadIdx.x >> 5] = wsum;
    __syncthreads();
    if (threadIdx.x == 0) {
        float t = 0.0f;
#pragma unroll
        for (int j = 0; j < NWAVES; ++j) t += smem[j];
        ws[blockIdx.x] = t;
    }
}

__global__ void __launch_bounds__(BLOCK)
angle_finalize_kernel(const float* __restrict__ ws, float* __restrict__ out) {
    // 1024 partials, 256 threads: one b128 load each.
    const v4f v = ((const v4f*)ws)[threadIdx.x];
    float acc = (v[0] + v[1]) + (v[2] + v[3]);

    __shared__ float smem[NWAVES];
    const float wsum = wave_sum_wmma(acc);
    if ((threadIdx.x & 31) == 0) smem[threadIdx.x >> 5] = wsum;
    __syncthreads();
    if (threadIdx.x == 0) {
        float t = 0.0f;
#pragma unroll
        for (int j = 0; j < NWAVES; ++j) t += smem[j];
        // mean(angle_rad) -> degrees -> /100
        out[0] = (float)((double)t * (180.0 / PI_D) / (double)NPIX / 100.0);
    }
}

extern "C" void kernel_launch(void* const* d_in, const int* in_sizes, int n_in,
                              void* d_out, int out_size, void* d_ws, size_t ws_size,
                              hipStream_t stream) {
    const float* x   = (const float*)d_in[0];
    const float* y   = (const float*)d_in[1];
    float*       ws  = (float*)d_ws;    // GRID1 floats = 4 KB, fully rewritten each call
    float*       out = (float*)d_out;

    angle_partial_kernel<<<GRID1, BLOCK, 0, stream>>>(x, y, ws);
    angle_finalize_kernel<<<1, BLOCK, 0, stream>>>(ws, out);
}